// DynamicElementAggregator_68015102100196
// MI455X (gfx1250) — compile-verified
//
#include <hip/hip_runtime.h>
#include <math.h>

typedef __attribute__((ext_vector_type(2))) float v2f;
typedef __attribute__((ext_vector_type(4))) float v4f;
typedef __attribute__((ext_vector_type(8))) float v8f;

#define B_SZ 2
#define N_SZ 1024
#define M_SZ 1024
#define D_SZ 768
#define F_SZ 384
#define H_SZ 72

#define XS_STRIDE 772   /* 768 + 4 pad: lane L -> bank 4L, conflict-free A-frag reads */
#define HS_STRIDE 388   /* 384 + 4 pad */

__device__ __forceinline__ float gelu_exact(float v) {
    return 0.5f * v * (1.0f + erff(v * 0.70710678118654752440f));
}

// ---------------------------------------------------------------------------
// Kernel A: router MLP on the WMMA pipe (fp32 WMMA, exact math).
// One 4-wave block per 16 output rows. rw[row][h] written to workspace.
// ---------------------------------------------------------------------------
__global__ __launch_bounds__(128) void router_mlp_wmma(
    const float* __restrict__ x,  const float* __restrict__ W1,
    const float* __restrict__ b1, const float* __restrict__ W2,
    const float* __restrict__ b2, float* __restrict__ rw)
{
    __shared__ float Xs[16 * XS_STRIDE];   // 16 x 768 X tile (padded)
    __shared__ float Hs[16 * HS_STRIDE];   // 16 x 384 hidden tile (padded)

    const int tid  = threadIdx.x;
    const int wave = tid >> 5;
    const int lane = tid & 31;
    const int l16  = lane & 15;
    const int hi   = lane >> 4;      // 0: lanes 0-15, 1: lanes 16-31
    const int koff = hi * 2;         // K offset inside a 4-wide k-step (ISA 16x4 A layout)
    const int row0 = blockIdx.x * 16;

    // Cooperative stage of X tile [16 x 768] into LDS (float4 per transfer)
    for (int i = tid; i < 16 * (D_SZ / 4); i += 128) {
        int r = i / (D_SZ / 4);
        int c = (i % (D_SZ / 4)) * 4;
        v4f v = *(const v4f*)(x + (size_t)(row0 + r) * D_SZ + c);
        *(v4f*)&Xs[r * XS_STRIDE + c] = v;
    }
    __syncthreads();

    // GEMM1 + exact GELU: Hs = gelu(X @ W1 + b1), 24 N-tiles split across 4 waves
    for (int t = wave; t < F_SZ / 16; t += 4) {
        v8f acc = {};
        const int colB = t * 16 + l16;
        for (int k = 0; k < D_SZ / 4; ++k) {
            const int k4 = k * 4;
            v2f a = *(const v2f*)&Xs[l16 * XS_STRIDE + k4 + koff];
            v2f b;
            b.x = W1[(size_t)(k4 + koff)     * F_SZ + colB];
            b.y = W1[(size_t)(k4 + koff + 1) * F_SZ + colB];
            acc = __builtin_amdgcn_wmma_f32_16x16x4_f32(
                false, a, false, b, (short)0, acc, false, false);
        }
        const float bb = b1[colB];
#pragma unroll
        for (int r = 0; r < 8; ++r) {                 // C layout: VGPR r -> M=r (+8 for hi lanes)
            int m = r + hi * 8;
            Hs[m * HS_STRIDE + colB] = gelu_exact(acc[r] + bb);
        }
    }
    __syncthreads();

    // GEMM2: rw = Hs @ W2 + b2  (N=72 -> 5 tiles, last tile column-clamped;
    // clamp keeps EXEC all-1s through the WMMAs, garbage cols >=72 never stored)
    for (int t = wave; t < 5; t += 4) {
        v8f acc = {};
        const int col  = t * 16 + l16;
        const int colc = (col < H_SZ) ? col : (H_SZ - 1);
        for (int k = 0; k < F_SZ / 4; ++k) {
            const int k4 = k * 4;
            v2f a = *(const v2f*)&Hs[l16 * HS_STRIDE + k4 + koff];
            v2f b;
            b.x = W2[(size_t)(k4 + koff)     * H_SZ + colc];
            b.y = W2[(size_t)(k4 + koff + 1) * H_SZ + colc];
            acc = __builtin_amdgcn_wmma_f32_16x16x4_f32(
                false, a, false, b, (short)0, acc, false, false);
        }
        if (col < H_SZ) {
            const float bb = b2[col];
#pragma unroll
            for (int r = 0; r < 8; ++r) {
                int m = r + hi * 8;
                rw[(size_t)(row0 + m) * H_SZ + col] = acc[r] + bb;
            }
        }
    }
}

// ---------------------------------------------------------------------------
// Kernel B: out[b,n,m] = sum_h rw[b,n,h] * scores[b,h,n,m] + bias
// Pure HBM streaming of 604 MB -> non-temporal float4 loads, one block per (b,n).
// ---------------------------------------------------------------------------
__global__ __launch_bounds__(256) void combine_heads(
    const float* __restrict__ scores, const float* __restrict__ rw,
    const float* __restrict__ bias,   float* __restrict__ out)
{
    __shared__ float w[H_SZ];
    const int bn = blockIdx.x;               // b*N + n
    const int t  = threadIdx.x;
    if (t < H_SZ) w[t] = rw[(size_t)bn * H_SZ + t];
    __syncthreads();

    const int b = bn >> 10;
    const int n = bn & (N_SZ - 1);
    const size_t hstride = (size_t)N_SZ * M_SZ;
    const float* p = scores + ((size_t)b * H_SZ * N_SZ + (size_t)n) * M_SZ + 4 * t;

    v4f acc = {};
#pragma unroll 4
    for (int h = 0; h < H_SZ; ++h) {
        v4f s = __builtin_nontemporal_load((const v4f*)(p + (size_t)h * hstride));
        acc += w[h] * s;
    }
    acc += bias[0];
    *(v4f*)(out + (size_t)bn * M_SZ + 4 * t) = acc;
}

extern "C" void kernel_launch(void* const* d_in, const int* in_sizes, int n_in,
                              void* d_out, int out_size, void* d_ws, size_t ws_size,
                              hipStream_t stream) {
    const float* x    = (const float*)d_in[0];
    const float* sc   = (const float*)d_in[1];
    const float* W1   = (const float*)d_in[2];
    const float* b1   = (const float*)d_in[3];
    const float* W2   = (const float*)d_in[4];
    const float* b2   = (const float*)d_in[5];
    const float* bias = (const float*)d_in[6];
    float* rw  = (float*)d_ws;     // [B*N, 72] = 576 KB scratch
    float* out = (float*)d_out;

    router_mlp_wmma<<<(B_SZ * N_SZ) / 16, 128, 0, stream>>>(x, W1, b1, W2, b2, rw);
    combine_heads<<<B_SZ * N_SZ, 256, 0, stream>>>(sc, rw, bias, out);
}